// FixedPastCausalAttention_57569741636010
// MI455X (gfx1250) — compile-verified
//
#include <hip/hip_runtime.h>
#include <hip/hip_bf16.h>

// ---------------------------------------------------------------------------
// FixedPastCausalAttention for MI455X (gfx1250, wave32, WMMA).
//
// Key observation: einsum('nhql,nlhd->nlhd') reduces over q, so the "attention"
// output is  colsum[n,h,l] * v_proj[n,l,h,d]  with colsum = per-column sum of
// the banded softmax scores. We therefore never materialize full scores x V;
// we only need column sums of the band softmax. All matmuls (q/k/v 64x64
// projections, banded q.k^T energies, and the dominant 1024x1024 output
// projection) go through V_WMMA_F32_16X16X4_F32 to keep exact fp32 numerics.
// ---------------------------------------------------------------------------

#define N_B   4
#define SEQ   1024
#define DIM   1024
#define NH    16
#define HD    64
#define HISTW 128
#define INV_SCALE 0.03125f   // 1 / sqrt(1024)

typedef float v2f __attribute__((ext_vector_type(2)));
typedef float v8f __attribute__((ext_vector_type(8)));

__device__ __forceinline__ v8f wmma_f32(v2f a, v2f b, v8f c) {
  // (neg_a, A, neg_b, B, c_mod, C, reuse_a, reuse_b)
  return __builtin_amdgcn_wmma_f32_16x16x4_f32(false, a, false, b, (short)0, c,
                                               false, false);
}

// Fragment layouts (f32, 16x16x4):
//   A (16x4): lane l, reg i -> A[l&15][i + (l>=16 ? 2 : 0)]
//   B (4x16): lane l, reg i -> B[i + (l>=16 ? 2 : 0)][l&15]
//   C (16x16): lane l, reg j -> C[j + (l>=16 ? 8 : 0)][l&15]

__global__ void zero_f32(float* __restrict__ p, int n) {
  int i = blockIdx.x * blockDim.x + threadIdx.x;
  if (i < n) p[i] = 0.0f;
}

// out = X_tile(64x64) @ W^T (nn.Linear), per (n, h, 64-row tile).
// outIsA=0 -> dst layout [nh][l][64]; outIsA=1 -> dst layout [n*SEQ+l][DIM] at col h*64.
__global__ void proj_kernel(const float* __restrict__ src,
                            const float* __restrict__ W,
                            float* __restrict__ dst, int outIsA) {
  __shared__ float sW[64][65];
  __shared__ float sX[64][65];
  const int tid  = threadIdx.x;            // 128 threads = 4 waves
  const int bid  = blockIdx.x;             // (n*NH + h)*16 + tile
  const int tile = bid & 15;
  const int nh   = bid >> 4;
  const int h    = nh & (NH - 1);
  const int n    = nh >> 4;
  const int l0   = tile * 64;

  const float* xbase = src + ((size_t)(n * SEQ + l0)) * DIM + h * HD;
  for (int idx = tid; idx < 64 * 64; idx += 128) {
    int r = idx >> 6, c = idx & 63;
    sW[r][c] = W[idx];
    sX[r][c] = xbase[(size_t)r * DIM + c];
  }
  __syncthreads();

  const int lane = tid & 31;
  const int mi   = tid >> 5;               // wave -> 16-row tile
  const int ln   = lane & 15;
  const int hi   = (lane >> 4) & 1;
  const int ko   = hi ? 2 : 0;

  v8f acc[4];
  for (int t = 0; t < 4; ++t)
    for (int j = 0; j < 8; ++j) acc[t][j] = 0.0f;

  for (int k0 = 0; k0 < 64; k0 += 4) {
    v2f a;
    a[0] = sX[mi * 16 + ln][k0 + ko];
    a[1] = sX[mi * 16 + ln][k0 + ko + 1];
#pragma unroll
    for (int ni = 0; ni < 4; ++ni) {
      v2f b;  // B[kk][nn] = W[ni*16+nn][k0+kk]  (x @ W^T)
      b[0] = sW[ni * 16 + ln][k0 + ko];
      b[1] = sW[ni * 16 + ln][k0 + ko + 1];
      acc[ni] = wmma_f32(a, b, acc[ni]);
    }
  }

#pragma unroll
  for (int ni = 0; ni < 4; ++ni)
    for (int j = 0; j < 8; ++j) {
      int row = mi * 16 + j + (hi ? 8 : 0);
      int col = ni * 16 + ln;
      size_t off = outIsA
          ? ((size_t)(n * SEQ + l0 + row)) * DIM + h * HD + col
          : ((size_t)nh * SEQ + l0 + row) * HD + col;
      dst[off] = acc[ni][j];
    }
}

// Banded energy + softmax + column-sum accumulation.
// One block per (n, h, 64-row q tile). Column window: [q0-128, q0+63] (192).
__global__ void band_kernel(const float* __restrict__ qp,
                            const float* __restrict__ kp,
                            float* __restrict__ colsum) {
  __shared__ float sc[64][193];
  const int tid = threadIdx.x;             // 128 threads = 4 waves
  const int bid = blockIdx.x;              // nh*16 + qt
  const int qt  = bid & 15;
  const int nh  = bid >> 4;
  const int q0  = qt * 64;
  const int w0  = q0 - 128;                // window start column (may be <0)
  const float* qb = qp + (size_t)nh * SEQ * HD;
  const float* kb = kp + (size_t)nh * SEQ * HD;

  const int lane = tid & 31;
  const int mi   = tid >> 5;
  const int ln   = lane & 15;
  const int hi   = (lane >> 4) & 1;
  const int ko   = hi ? 2 : 0;

  v8f acc[12];
  for (int t = 0; t < 12; ++t)
    for (int j = 0; j < 8; ++j) acc[t][j] = 0.0f;

  for (int k0 = 0; k0 < 64; k0 += 4) {
    const int qrow = q0 + mi * 16 + ln;
    v2f a;
    a[0] = qb[(size_t)qrow * HD + k0 + ko];
    a[1] = qb[(size_t)qrow * HD + k0 + ko + 1];
#pragma unroll
    for (int ni = 0; ni < 12; ++ni) {
      const int c = w0 + ni * 16 + ln;     // key column (lane-dependent)
      v2f b;
      if (c >= 0) {                        // reconverges before the WMMA
        b[0] = kb[(size_t)c * HD + k0 + ko];
        b[1] = kb[(size_t)c * HD + k0 + ko + 1];
      } else {
        b[0] = 0.0f; b[1] = 0.0f;
      }
      acc[ni] = wmma_f32(a, b, acc[ni]);
    }
  }

#pragma unroll
  for (int ni = 0; ni < 12; ++ni)
    for (int j = 0; j < 8; ++j)
      sc[mi * 16 + j + (hi ? 8 : 0)][ni * 16 + ln] = acc[ni][j];
  __syncthreads();

  // Row softmax over the valid band: j in [max(r+1, -w0), r+128].
  if (tid < 64) {
    const int r = tid;
    int jlo = r + 1;
    if (-w0 > jlo) jlo = -w0;
    const int jhi = r + 128;               // <= 191
    float m = -3.0e38f;
    for (int j = jlo; j <= jhi; ++j) m = fmaxf(m, sc[r][j]);
    float s = 0.0f;
    for (int j = jlo; j <= jhi; ++j) {
      float e = __expf((sc[r][j] - m) * INV_SCALE);
      sc[r][j] = e;
      s += e;
    }
    const float inv = 1.0f / s;
    for (int j = 0; j < 192; ++j)
      sc[r][j] = (j >= jlo && j <= jhi) ? sc[r][j] * inv : 0.0f;
  }
  __syncthreads();

  // Partial column sums over this block's 64 q-rows -> global atomic add.
  for (int j = tid; j < 192; j += 128) {
    const int c = w0 + j;
    if (c >= 0) {
      float s = 0.0f;
      for (int r = 0; r < 64; ++r) s += sc[r][j];
      atomicAdd(&colsum[(size_t)nh * SEQ + c], s);
    }
  }
}

// A[n*SEQ+l][h*64+d] *= colsum[n*NH+h][l]
__global__ void scale_kernel(float* __restrict__ A,
                             const float* __restrict__ colsum) {
  const size_t i = (size_t)blockIdx.x * blockDim.x + threadIdx.x; // < 2^22
  const int n   = (int)(i >> 20);          // SEQ*DIM = 2^20
  const int rem = (int)(i & 0xFFFFF);
  const int l   = rem >> 10;
  const int h   = (rem & 1023) >> 6;
  A[i] *= colsum[((size_t)(n * NH + h)) * SEQ + l];
}

// out[M=4096][1024] = A @ Wo^T + bo.  Block tile 128x64, LDS k-chunks of 64.
__global__ void gemm_kernel(const float* __restrict__ A,
                            const float* __restrict__ Wo,
                            const float* __restrict__ bo,
                            float* __restrict__ out) {
  __shared__ float sA[128][65];
  __shared__ float sB[64][65];
  const int tid  = threadIdx.x;            // 256 threads = 8 waves
  const int m0   = blockIdx.x * 128;
  const int nb0  = blockIdx.y * 64;
  const int lane = tid & 31;
  const int mi   = tid >> 5;               // wave -> 16-row tile (0..7)
  const int ln   = lane & 15;
  const int hi   = (lane >> 4) & 1;
  const int ko   = hi ? 2 : 0;

  v8f acc[4];
  for (int t = 0; t < 4; ++t)
    for (int j = 0; j < 8; ++j) acc[t][j] = 0.0f;

  for (int kc = 0; kc < DIM; kc += 64) {
    for (int idx = tid; idx < 128 * 64; idx += 256) {
      int r = idx >> 6, c = idx & 63;
      sA[r][c] = A[(size_t)(m0 + r) * DIM + kc + c];
    }
    for (int idx = tid; idx < 64 * 64; idx += 256) {
      int r = idx >> 6, c = idx & 63;
      sB[r][c] = Wo[(size_t)(nb0 + r) * DIM + kc + c];
    }
    __syncthreads();
#pragma unroll 4
    for (int kk = 0; kk < 64; kk += 4) {
      v2f a;
      a[0] = sA[mi * 16 + ln][kk + ko];
      a[1] = sA[mi * 16 + ln][kk + ko + 1];
#pragma unroll
      for (int ni = 0; ni < 4; ++ni) {
        v2f b;  // B[kk][nn] = Wo[nb0+ni*16+nn][kc+kk]
        b[0] = sB[ni * 16 + ln][kk + ko];
        b[1] = sB[ni * 16 + ln][kk + ko + 1];
        acc[ni] = wmma_f32(a, b, acc[ni]);
      }
    }
    __syncthreads();
  }

#pragma unroll
  for (int ni = 0; ni < 4; ++ni) {
    const int col  = nb0 + ni * 16 + ln;
    const float bb = bo[col];
    for (int j = 0; j < 8; ++j) {
      const int row = m0 + mi * 16 + j + (hi ? 8 : 0);
      out[(size_t)row * DIM + col] = acc[ni][j] + bb;
    }
  }
}

extern "C" void kernel_launch(void* const* d_in, const int* in_sizes, int n_in,
                              void* d_out, int out_size, void* d_ws, size_t ws_size,
                              hipStream_t stream) {
  const float* values = (const float*)d_in[0];
  const float* keys   = (const float*)d_in[1];
  const float* query  = (const float*)d_in[2];
  // d_in[3] is the mask: deterministically the causal band; computed analytically.
  const float* Wv = (const float*)d_in[4];
  const float* Wk = (const float*)d_in[5];
  const float* Wq = (const float*)d_in[6];
  const float* Wo = (const float*)d_in[7];
  const float* bo = (const float*)d_in[8];
  float* out = (float*)d_out;

  char*  ws     = (char*)d_ws;
  float* qp     = (float*)(ws);                               // 16 MB
  float* kp     = (float*)(ws + (size_t)16 * 1024 * 1024);    // 16 MB
  float* Amat   = (float*)(ws + (size_t)32 * 1024 * 1024);    // 16 MB
  float* colsum = (float*)(ws + (size_t)48 * 1024 * 1024);    // 256 KB

  zero_f32<<<(N_B * NH * SEQ) / 256, 256, 0, stream>>>(colsum, N_B * NH * SEQ);
  proj_kernel<<<N_B * NH * (SEQ / 64), 128, 0, stream>>>(query,  Wq, qp,   0);
  proj_kernel<<<N_B * NH * (SEQ / 64), 128, 0, stream>>>(keys,   Wk, kp,   0);
  proj_kernel<<<N_B * NH * (SEQ / 64), 128, 0, stream>>>(values, Wv, Amat, 1);
  band_kernel<<<N_B * NH * (SEQ / 64), 128, 0, stream>>>(qp, kp, colsum);
  scale_kernel<<<(N_B * SEQ * DIM) / 256, 256, 0, stream>>>(Amat, colsum);
  gemm_kernel<<<dim3((N_B * SEQ) / 128, DIM / 64), 256, 0, stream>>>(Amat, Wo, bo, out);
}